// TaperingLayer_42880953483791
// MI455X (gfx1250) — compile-verified
//
#include <hip/hip_runtime.h>
#include <math.h>

// ---------------------------------------------------------------------------
// TaperingLayer reduced analytically to:  out[b] = M * patch_b * M^T
//   patch_b = img[b, x_pos:x_pos+33, y_pos:y_pos+33]   (33x33 real)
//   M       = fixed 33x33 complex matrix (closed form, see derivation)
// Implemented with V_WMMA_F32_16X16X4_F32 on 48x48 zero-padded tiles.
// ---------------------------------------------------------------------------

typedef __attribute__((ext_vector_type(2))) float v2f;
typedef __attribute__((ext_vector_type(8))) float v8f;

#define NP   48      // padded tile dim (3 x 16)
#define PT   33      // P
#define NWIN 479     // WIN

__device__ __forceinline__ v8f wmma4(v2f a, v2f b, v8f c) {
    // D = A(16x4, f32) * B(4x16, f32) + C(16x16, f32)
    return __builtin_amdgcn_wmma_f32_16x16x4_f32(
        /*neg_a=*/false, a, /*neg_b=*/false, b,
        /*c_mod=*/(short)0, c, /*reuse_a=*/false, /*reuse_b=*/false);
}

// Build M (row-major [a][m], zero-padded to 48x48) into workspace.
__global__ void make_M_kernel(float* __restrict__ Mr, float* __restrict__ Mi) {
    int idx = blockIdx.x * blockDim.x + threadIdx.x;
    if (idx >= NP * NP) return;
    int a = idx / NP, m = idx % NP;
    float re = 0.f, im = 0.f;
    if (a < PT && m < PT) {
        int sa = (a + 16) % PT;                 // final ifftshift folded in
        float c = (float)sa * 14.9375f;         // linspace(0, 478, 33): step = 239/16 exact
        int i0 = (int)floorf(c);
        if (i0 > NWIN - 2) i0 = NWIN - 2;       // clip (sa==32 -> i0=477, f=1)
        float f = c - (float)i0;
        int t = m - 16;                         // patch offset from window center
        int k1 = (int)(((long long)t * (long long)(i0 + 1)) % NWIN); if (k1 < 0) k1 += NWIN;
        int k2 = (int)(((long long)t * (long long)(i0 + 2)) % NWIN); if (k2 < 0) k2 += NWIN;
        const float w = 6.28318530717958647692f / (float)NWIN;
        float s1, c1, s2, c2;
        sincosf(w * (float)k1, &s1, &c1);       // exp(-i*2*pi*k/N) = c - i*s
        sincosf(w * (float)k2, &s2, &c2);
        re =  (float)NWIN * ((1.f - f) * c1 + f * c2);
        im = -(float)NWIN * ((1.f - f) * s1 + f * s2);
    }
    Mr[idx] = re;
    Mi[idx] = im;
}

// One block (4 waves) per image; b<64 -> ref, else mov. Output layout matches
// concatenated (feature1, feature2) flat: out + b*33*33*2.
__global__ void tapering_kernel(const float* __restrict__ ref,
                                const float* __restrict__ mov,
                                const int*   __restrict__ xpos,
                                const int*   __restrict__ ypos,
                                const float* __restrict__ gMr,
                                const float* __restrict__ gMi,
                                float* __restrict__ out) {
    __shared__ float sP [NP][NP];   // real patch
    __shared__ float sMr[NP][NP];   // Re(M)
    __shared__ float sMi[NP][NP];   // Im(M)
    __shared__ float sTr[NP][NP];   // Re(T) = P * Mr^T
    __shared__ float sTi[NP][NP];   // Im(T) = P * Mi^T

    const int b   = blockIdx.x;     // 0..127
    const int tid = threadIdx.x;    // 0..127
    const int xp = xpos[0];
    const int yp = ypos[0];
    const float* img = (b < 64) ? (ref + (size_t)b * 256 * 256)
                                : (mov + (size_t)(b - 64) * 256 * 256);

    for (int idx = tid; idx < NP * NP; idx += 128) {
        int r = idx / NP, c = idx % NP;
        sMr[r][c] = gMr[idx];
        sMi[r][c] = gMi[idx];
        float v = 0.f;
        if (r < PT && c < PT) v = img[(size_t)(xp + r) * 256 + (yp + c)];
        sP[r][c] = v;
    }
    __syncthreads();

    const int lane = tid & 31;
    const int wave = tid >> 5;
    const int n = lane & 15;        // A-row / B-col / D-col index
    const int g = lane >> 4;        // half-wave select

    // ---- Stage 1: T = patch * M^T  (two real 48x48x48 GEMMs) ----
    for (int t = wave; t < 9; t += 4) {
        const int ti = t / 3, tj = t % 3;
        v8f accr = {}; v8f acci = {};
        for (int k0 = 0; k0 < NP; k0 += 4) {
            v2f a, br, bi;
            // A frag: A[m][k0+2g], A[m][k0+2g+1]
            a.x  = sP [ti * 16 + n][k0 + 2 * g];
            a.y  = sP [ti * 16 + n][k0 + 2 * g + 1];
            // B = M^T  =>  B[k][ncol] = M[ncol][k]
            br.x = sMr[tj * 16 + n][k0 + 2 * g];
            br.y = sMr[tj * 16 + n][k0 + 2 * g + 1];
            bi.x = sMi[tj * 16 + n][k0 + 2 * g];
            bi.y = sMi[tj * 16 + n][k0 + 2 * g + 1];
            accr = wmma4(a, br, accr);
            acci = wmma4(a, bi, acci);
        }
        for (int v = 0; v < 8; ++v) {
            sTr[ti * 16 + v + 8 * g][tj * 16 + n] = accr[v];
            sTi[ti * 16 + v + 8 * g][tj * 16 + n] = acci[v];
        }
    }
    __syncthreads();

    // ---- Stage 2: O = M * T  (complex), write interleaved re/im ----
    float* outb = out + (size_t)b * (PT * PT * 2);
    for (int t = wave; t < 9; t += 4) {
        const int ti = t / 3, tj = t % 3;
        v8f accr = {}; v8f acci = {};
        for (int k0 = 0; k0 < NP; k0 += 4) {
            v2f ar, ai, nai, br, bi;
            ar.x  = sMr[ti * 16 + n][k0 + 2 * g];
            ar.y  = sMr[ti * 16 + n][k0 + 2 * g + 1];
            ai.x  = sMi[ti * 16 + n][k0 + 2 * g];
            ai.y  = sMi[ti * 16 + n][k0 + 2 * g + 1];
            nai.x = -ai.x; nai.y = -ai.y;        // f32 NEG bits are C-only; negate in VALU
            br.x  = sTr[k0 + 2 * g][tj * 16 + n];
            br.y  = sTr[k0 + 2 * g + 1][tj * 16 + n];
            bi.x  = sTi[k0 + 2 * g][tj * 16 + n];
            bi.y  = sTi[k0 + 2 * g + 1][tj * 16 + n];
            accr = wmma4(ar,  br, accr);         // + Mr*Tr
            accr = wmma4(nai, bi, accr);         // - Mi*Ti
            acci = wmma4(ar,  bi, acci);         // + Mr*Ti
            acci = wmma4(ai,  br, acci);         // + Mi*Tr
        }
        for (int v = 0; v < 8; ++v) {
            int row = ti * 16 + v + 8 * g;
            int col = tj * 16 + n;
            if (row < PT && col < PT) {
                outb[((size_t)row * PT + col) * 2 + 0] = accr[v];
                outb[((size_t)row * PT + col) * 2 + 1] = acci[v];
            }
        }
    }
}

extern "C" void kernel_launch(void* const* d_in, const int* in_sizes, int n_in,
                              void* d_out, int out_size, void* d_ws, size_t ws_size,
                              hipStream_t stream) {
    const float* ref = (const float*)d_in[0];   // (64, 256, 256) f32
    const float* mov = (const float*)d_in[1];   // (64, 256, 256) f32
    const int*   xp  = (const int*)d_in[2];     // scalar
    const int*   yp  = (const int*)d_in[3];     // scalar

    float* Mr = (float*)d_ws;                   // 48*48 floats
    float* Mi = Mr + NP * NP;                   // 48*48 floats (total 18 KB of ws)

    make_M_kernel<<<(NP * NP + 255) / 256, 256, 0, stream>>>(Mr, Mi);
    tapering_kernel<<<128, 128, 0, stream>>>(ref, mov, xp, yp, Mr, Mi, (float*)d_out);
}